// GAT_72662256714420
// MI455X (gfx1250) — compile-verified
//
#include <hip/hip_runtime.h>
#include <math.h>

typedef __attribute__((ext_vector_type(2))) float v2f;
typedef __attribute__((ext_vector_type(8))) float v8f;

#define N_NODES 50000
#define N_EDGES 800000
#define IN_F    256
#define HEADS   4
#define CH      64
#define HC      (HEADS * CH)   // 256

// ---------------------------------------------------------------------------
// GEMM: C[M x 256] = A[M x K] * B[K x 256], exact f32 via V_WMMA_F32_16X16X4_F32.
// Each wave32 computes a 16x64 output strip (4 tiles, 4 independent
// accumulators -> 4 overlapping WMMA chains, 4x A-fragment reuse).
// 8 waves (256 threads) per block.
// ---------------------------------------------------------------------------
template<int K>
__global__ __launch_bounds__(256)
void wmma_gemm_kernel(const float* __restrict__ A, const float* __restrict__ B,
                      float* __restrict__ C, int M)
{
    const int lane   = threadIdx.x & 31;
    const int wave   = blockIdx.x * 8 + (threadIdx.x >> 5);
    const int mTile  = wave >> 2;                 // 16-row strip
    const int nGroup = wave & 3;                  // 64-col strip (4 x 16)
    if (mTile * 16 >= M) return;                  // wave-uniform guard

    const int half = lane >> 4;                   // 0: K={0,1}, 1: K={2,3}
    const int l    = lane & 15;
    const long rowBase = (long)(mTile * 16 + l) * K;   // A row for this lane
    const int  col0    = nGroup * 64 + l;              // first B column

    v8f acc0 = {}, acc1 = {}, acc2 = {}, acc3 = {};
    #pragma unroll 2
    for (int k = 0; k < K; k += 4) {
        const int ka = k + half * 2;                   // even -> 8B aligned
        v2f a = *(const v2f*)(A + rowBase + ka);       // A[M][ka..ka+1]
        __builtin_prefetch(A + rowBase + ka + 32, 0, 3);   // near-cache prefetch
        const float* Bp = B + (long)ka * HC + col0;
        v2f b0, b1, b2, b3;
        b0.x = Bp[0];  b0.y = Bp[HC];                  // B[ka][col], B[ka+1][col]
        b1.x = Bp[16]; b1.y = Bp[HC + 16];
        b2.x = Bp[32]; b2.y = Bp[HC + 32];
        b3.x = Bp[48]; b3.y = Bp[HC + 48];
        acc0 = __builtin_amdgcn_wmma_f32_16x16x4_f32(false, a, false, b0, (short)0, acc0, false, false);
        acc1 = __builtin_amdgcn_wmma_f32_16x16x4_f32(false, a, false, b1, (short)0, acc1, false, false);
        acc2 = __builtin_amdgcn_wmma_f32_16x16x4_f32(false, a, false, b2, (short)0, acc2, false, false);
        acc3 = __builtin_amdgcn_wmma_f32_16x16x4_f32(false, a, false, b3, (short)0, acc3, false, false);
    }

    // C layout per tile: VGPR i -> M = mTile*16 + half*8 + i, N = col
    float* Cp = C + (long)(mTile * 16 + half * 8) * HC + nGroup * 64 + l;
    #pragma unroll
    for (int i = 0; i < 8; ++i) {
        Cp[(long)i * HC +  0] = acc0[i];
        Cp[(long)i * HC + 16] = acc1[i];
        Cp[(long)i * HC + 32] = acc2[i];
        Cp[(long)i * HC + 48] = acc3[i];
    }
}

// ---------------------------------------------------------------------------
// Per-node attention logits: alpha_s/d[n][h] = sum_c h[n][h][c] * a_{s,d}[h][c]
// ---------------------------------------------------------------------------
__global__ __launch_bounds__(256)
void alpha_kernel(const float* __restrict__ h, const float* __restrict__ a_src,
                  const float* __restrict__ a_dst, float* __restrict__ as,
                  float* __restrict__ ad, int n)
{
    int gid = blockIdx.x * blockDim.x + threadIdx.x;
    if (gid >= n * HEADS) return;
    int node = gid >> 2, hd = gid & 3;
    const float* hp = h + (long)node * HC + hd * CH;
    const float* s  = a_src + hd * CH;
    const float* d  = a_dst + hd * CH;
    float ss = 0.f, dd = 0.f;
    #pragma unroll 8
    for (int c = 0; c < CH; ++c) { float v = hp[c]; ss += v * s[c]; dd += v * d[c]; }
    as[gid] = ss;
    ad[gid] = dd;
}

// Monotone float <-> uint map so segment-max runs on native GLOBAL_ATOMIC_MAX_U32.
__device__ __forceinline__ unsigned fmap(float x) {
    unsigned b = __float_as_uint(x);
    return (b & 0x80000000u) ? ~b : (b | 0x80000000u);
}
__device__ __forceinline__ float funmap(unsigned u) {
    unsigned b = (u & 0x80000000u) ? (u ^ 0x80000000u) : ~u;
    return __uint_as_float(b);
}

__device__ __forceinline__ void edge_endpoints(const int* __restrict__ ei, int e,
                                               int nE, int& src, int& dst) {
    if (e < nE) { src = ei[e]; dst = ei[nE + e]; }
    else        { src = dst = e - nE; }               // appended self-loops
}

// Pass 1: segment max of leaky_relu(alpha_s[src] + alpha_d[dst]) per head.
__global__ __launch_bounds__(256)
void edge_max_kernel(const int* __restrict__ ei, const float* __restrict__ as,
                     const float* __restrict__ ad, unsigned* __restrict__ m,
                     int nE, int nN)
{
    int e = blockIdx.x * blockDim.x + threadIdx.x;
    if (e >= nE + nN) return;
    int src, dst; edge_endpoints(ei, e, nE, src, dst);
    #pragma unroll
    for (int hd = 0; hd < HEADS; ++hd) {
        float v = as[src * HEADS + hd] + ad[dst * HEADS + hd];
        v = (v > 0.f) ? v : 0.2f * v;
        atomicMax(&m[dst * HEADS + hd], fmap(v));
    }
}

// Pass 2: segment sum of exp(e - max[dst]).
__global__ __launch_bounds__(256)
void edge_denom_kernel(const int* __restrict__ ei, const float* __restrict__ as,
                       const float* __restrict__ ad, const unsigned* __restrict__ m,
                       float* __restrict__ denom, int nE, int nN)
{
    int e = blockIdx.x * blockDim.x + threadIdx.x;
    if (e >= nE + nN) return;
    int src, dst; edge_endpoints(ei, e, nE, src, dst);
    #pragma unroll
    for (int hd = 0; hd < HEADS; ++hd) {
        float v = as[src * HEADS + hd] + ad[dst * HEADS + hd];
        v = (v > 0.f) ? v : 0.2f * v;
        float w = __expf(v - funmap(m[dst * HEADS + hd]));
        atomicAdd(&denom[dst * HEADS + hd], w);
    }
}

// Pass 3: one block per edge; 256 lanes cover (head, channel); coef computed
// once per head in LDS; native f32 atomic adds land in the 192MB L2.
__global__ __launch_bounds__(256)
void aggregate_kernel(const int* __restrict__ ei, const float* __restrict__ h,
                      const float* __restrict__ as, const float* __restrict__ ad,
                      const unsigned* __restrict__ m, const float* __restrict__ denom,
                      float* __restrict__ accum, int nE, int nN)
{
    __shared__ float s_coef[HEADS];
    __shared__ int   s_sd[2];
    int e = blockIdx.x;
    if (threadIdx.x == 0) {
        int src, dst; edge_endpoints(ei, e, nE, src, dst);
        s_sd[0] = src; s_sd[1] = dst;
    }
    __syncthreads();
    const int src = s_sd[0], dst = s_sd[1];
    if (threadIdx.x < HEADS) {
        int hd  = threadIdx.x;
        float v = as[src * HEADS + hd] + ad[dst * HEADS + hd];
        v = (v > 0.f) ? v : 0.2f * v;
        float w = __expf(v - funmap(m[dst * HEADS + hd]));
        s_coef[hd] = w / (denom[dst * HEADS + hd] + 1e-16f);
    }
    __syncthreads();
    const int hd = threadIdx.x >> 6, c = threadIdx.x & 63;
    const float val = h[(long)src * HC + hd * CH + c] * s_coef[hd];
    atomicAdd(&accum[(long)dst * HC + hd * CH + c], val);
}

// Head-mean + bias (+ optional ELU for layer 1).
__global__ __launch_bounds__(256)
void finalize_kernel(const float* __restrict__ accum, const float* __restrict__ bias,
                     float* __restrict__ out, int n, int applyElu)
{
    int gid = blockIdx.x * blockDim.x + threadIdx.x;
    if (gid >= n * CH) return;
    int node = gid >> 6, c = gid & 63;
    const float* ap = accum + (long)node * HC + c;
    float v = 0.25f * (ap[0] + ap[CH] + ap[2 * CH] + ap[3 * CH]) + bias[c];
    if (applyElu) v = (v > 0.f) ? v : expm1f(v);
    out[gid] = v;
}

// ---------------------------------------------------------------------------
extern "C" void kernel_launch(void* const* d_in, const int* in_sizes, int n_in,
                              void* d_out, int out_size, void* d_ws, size_t ws_size,
                              hipStream_t stream)
{
    const float* x   = (const float*)d_in[0];
    const int*   ei  = (const int*)  d_in[1];
    const float* W1  = (const float*)d_in[2];
    const float* a1s = (const float*)d_in[3];
    const float* a1d = (const float*)d_in[4];
    const float* b1  = (const float*)d_in[5];
    const float* W2  = (const float*)d_in[6];
    const float* a2s = (const float*)d_in[7];
    const float* a2d = (const float*)d_in[8];
    const float* b2  = (const float*)d_in[9];
    float* out = (float*)d_out;

    // Workspace carve-up (~118.4 MB of f32; whole set fits in MI455X's 192MB L2)
    float*    ws    = (float*)d_ws;
    float*    h     = ws;                                   // [N,256]
    float*    as    = h     + (size_t)N_NODES * HC;         // [N,4]
    float*    ad    = as    + (size_t)N_NODES * HEADS;      // [N,4]
    unsigned* m     = (unsigned*)(ad + (size_t)N_NODES * HEADS); // [N,4] mapped max
    float*    denom = (float*)m + (size_t)N_NODES * HEADS;  // [N,4]
    float*    accum = denom + (size_t)N_NODES * HEADS;      // [N,256]
    float*    hmid  = accum + (size_t)N_NODES * HC;         // [N,64]

    const int EE = N_EDGES + N_NODES;                       // edges + self-loops
    const dim3 b256(256);
    const int gemmWaves   = (N_NODES / 16) * 4;             // 16-row x 64-col strips
    const int gemmBlocks  = (gemmWaves + 7) / 8;
    const int edgeBlocks  = (EE + 255) / 256;
    const int alphaBlocks = (N_NODES * HEADS + 255) / 256;
    const int finBlocks   = (N_NODES * CH + 255) / 256;

    // ----------------- Layer 1 -----------------
    hipMemsetAsync(m,     0, (size_t)N_NODES * HEADS * 4, stream);
    hipMemsetAsync(denom, 0, (size_t)N_NODES * HEADS * 4, stream);
    hipMemsetAsync(accum, 0, (size_t)N_NODES * HC * 4, stream);

    wmma_gemm_kernel<IN_F><<<gemmBlocks, b256, 0, stream>>>(x, W1, h, N_NODES);
    alpha_kernel<<<alphaBlocks, b256, 0, stream>>>(h, a1s, a1d, as, ad, N_NODES);
    edge_max_kernel<<<edgeBlocks, b256, 0, stream>>>(ei, as, ad, m, N_EDGES, N_NODES);
    edge_denom_kernel<<<edgeBlocks, b256, 0, stream>>>(ei, as, ad, m, denom, N_EDGES, N_NODES);
    aggregate_kernel<<<EE, b256, 0, stream>>>(ei, h, as, ad, m, denom, accum, N_EDGES, N_NODES);
    finalize_kernel<<<finBlocks, b256, 0, stream>>>(accum, b1, hmid, N_NODES, 1);

    // ----------------- Layer 2 -----------------
    hipMemsetAsync(m,     0, (size_t)N_NODES * HEADS * 4, stream);
    hipMemsetAsync(denom, 0, (size_t)N_NODES * HEADS * 4, stream);
    hipMemsetAsync(accum, 0, (size_t)N_NODES * HC * 4, stream);

    wmma_gemm_kernel<CH><<<gemmBlocks, b256, 0, stream>>>(hmid, W2, h, N_NODES);
    alpha_kernel<<<alphaBlocks, b256, 0, stream>>>(h, a2s, a2d, as, ad, N_NODES);
    edge_max_kernel<<<edgeBlocks, b256, 0, stream>>>(ei, as, ad, m, N_EDGES, N_NODES);
    edge_denom_kernel<<<edgeBlocks, b256, 0, stream>>>(ei, as, ad, m, denom, N_EDGES, N_NODES);
    aggregate_kernel<<<EE, b256, 0, stream>>>(ei, h, as, ad, m, denom, accum, N_EDGES, N_NODES);
    finalize_kernel<<<finBlocks, b256, 0, stream>>>(accum, b2, out, N_NODES, 0);
}